// ShiftNMF_74577812127777
// MI455X (gfx1250) — compile-verified
//
#include <hip/hip_runtime.h>

typedef __attribute__((ext_vector_type(2))) float v2f;
typedef __attribute__((ext_vector_type(8))) float v8f;
typedef unsigned int u32;
typedef __attribute__((ext_vector_type(4))) u32 v4u;
typedef __attribute__((ext_vector_type(8))) int v8i;
typedef __attribute__((ext_vector_type(4))) int v4i;

#define N_ROWS 1024
#define RANK   16
#define M_COLS 4096

#define TWO_PI 6.28318530717958647692f

// ---- workspace layout (bytes) ----
// [0,       262144)  : spH  = softplus(H), 16 x 4096 f32
// [262144,  786432)  : Hft  = full DFT of spH, m-major interleaved complex:
//                      float2 Hft[4096][16]  (re, im)
#define WS_SPH_OFF 0
#define WS_HFT_OFF 262144

__device__ __forceinline__ float softplus_f(float x) {
  if (x > 20.0f) return x;
  if (x < -20.0f) return __expf(x);
  return log1pf(__expf(x));
}

// ------------------------------------------------------------------
// Kernel 0: spH = softplus(H)
// ------------------------------------------------------------------
__global__ void k_softplus_h(const float* __restrict__ H, float* __restrict__ spH) {
  int i = blockIdx.x * blockDim.x + threadIdx.x;
  if (i < RANK * M_COLS) spH[i] = softplus_f(H[i]);
}

// ------------------------------------------------------------------
// Kernel 1: rank-16 DFT as f32 WMMA matmul.
// One wave (32 lanes) per 16-wide m-tile. Accumulates
//   Hr[d,m] = sum_k spH[d,k] * cos(2*pi*k*m/M)
//   Hi[d,m] = -sum_k spH[d,k] * sin(2*pi*k*m/M)
// using V_WMMA_F32_16X16X4_F32 with twiddles generated by a
// unit-modulus rotation recurrence (no sincos in the hot loop).
// ------------------------------------------------------------------
__global__ void __launch_bounds__(32) k_dft_wmma(const float* __restrict__ spH,
                                                 float* __restrict__ Hft) {
  const int lane = threadIdx.x;           // 0..31, full wave (EXEC all 1s)
  const int col  = lane & 15;             // m within tile
  const int m    = blockIdx.x * 16 + col; // global m
  const bool hi  = lane >= 16;            // lanes 16..31 own K rows k+2,k+3

  const float delta = TWO_PI * (float)m / (float)M_COLS; // phase step per k

  // B-matrix lane state: this lane holds rows (k+r0, k+r0+1), r0 = hi?2:0
  float a0 = hi ? 2.0f * delta : 0.0f;
  float a1 = hi ? 3.0f * delta : delta;
  float c0, s0, c1, s1, cr, sr;
  __sincosf(a0, &s0, &c0);
  __sincosf(a1, &s1, &c1);
  __sincosf(4.0f * delta, &sr, &cr);      // rotation per k += 4

  // A-matrix pointer: row d = lane&15, cols k + (hi?2:0) + {0,1}
  const float* aptr = spH + (size_t)(lane & 15) * M_COLS + (hi ? 2 : 0);

  v8f accC = {};  // cos accumulator -> Re(Hft)
  v8f accS = {};  // sin accumulator -> -Im(Hft)

  for (int k = 0; k < M_COLS; k += 4) {
    v2f a = *(const v2f*)(aptr + k);
    v2f bc; bc.x = c0; bc.y = c1;
    v2f bs; bs.x = s0; bs.y = s1;
    accC = __builtin_amdgcn_wmma_f32_16x16x4_f32(
        false, a, false, bc, (short)0, accC, false, false);
    accS = __builtin_amdgcn_wmma_f32_16x16x4_f32(
        false, a, false, bs, (short)0, accS, false, false);

    // advance twiddles: (c,s) *= (cr + i sr)
    float nc0 = fmaf(c0, cr, -s0 * sr);
    float ns0 = fmaf(s0, cr,  c0 * sr);
    float nc1 = fmaf(c1, cr, -s1 * sr);
    float ns1 = fmaf(s1, cr,  c1 * sr);
    c0 = nc0; s0 = ns0; c1 = nc1; s1 = ns1;

    if ((k & 255) == 252) {  // renormalize to kill drift (Newton 1/sqrt step)
      float r0 = 1.5f - 0.5f * (c0 * c0 + s0 * s0); c0 *= r0; s0 *= r0;
      float r1 = 1.5f - 0.5f * (c1 * c1 + s1 * s1); c1 *= r1; s1 *= r1;
    }
  }

  // C/D layout: VGPR j holds M=j (lanes 0-15) / M=j+8 (lanes 16-31), N=col.
  #pragma unroll
  for (int j = 0; j < 8; ++j) {
    int d = j + (hi ? 8 : 0);
    float2 v;
    v.x = accC[j];
    v.y = -accS[j];
    *(float2*)(Hft + ((size_t)m * RANK + d) * 2) = v;
  }
}

// ------------------------------------------------------------------
// Kernel 2: V[n,m] = sum_d spW[n,d] * e^{-i 2pi tau[n,d] m / M} * Hft[d,m]
// Block = 256 threads = 16 n-groups x 16 m-lanes. Each 32 KB Hft m-chunk
// is staged into LDS by the Tensor Data Mover (tensor_load_to_lds), with
// TDM hardware padding (32-DWORD interval, 2-DWORD pad) producing the
// bank-conflict-free stride-17 float2 row layout. Phase factors advanced
// by a rotation recurrence (sincos only at chunk seeds).
// ------------------------------------------------------------------
#define MC 256          // m-chunk length
#define LSTR 17         // padded LDS row stride in float2 (bank spread)

__global__ void __launch_bounds__(256) k_main(const float* __restrict__ W,
                                              const float* __restrict__ tau,
                                              const float* __restrict__ Hft,
                                              float* __restrict__ out) {
  __shared__ float2 tile[MC * LSTR];  // 34 KB

  const int tid = threadIdx.x;
  const int ng  = tid >> 4;               // 0..15 -> n within block
  const int tm  = tid & 15;               // 0..15 -> m lane
  const int n   = blockIdx.x * 16 + ng;

  const float k2pi = TWO_PI / (float)M_COLS;

  float spw[RANK], dph[RANK], cd[RANK], sd[RANK];
  #pragma unroll
  for (int d = 0; d < RANK; ++d) {
    spw[d] = softplus_f(W[(size_t)n * RANK + d]);
    float t = tau[(size_t)n * RANK + d];
    dph[d]  = k2pi * t;                       // phase per unit m
    __sincosf(dph[d] * 16.0f, &sd[d], &cd[d]); // rotation per 16-m stride
  }

  for (int m0 = 0; m0 < M_COLS; m0 += MC) {
    __syncthreads();  // previous chunk fully consumed before overwrite
    const float2* src = (const float2*)Hft + (size_t)m0 * RANK;

    // prefetch next chunk toward L2 while TDM fills this one
    if (m0 + MC < M_COLS)
      __builtin_prefetch((const char*)(src + (size_t)MC * RANK) + tid * 128, 0, 1);

    if (tid < 32) {  // wave 0 drives the Tensor Data Mover
      unsigned long long ga = (unsigned long long)(const void*)src;
      u32 lds_base = (u32)(unsigned long long)(void*)tile;  // LDS offset = addr[31:0]

      // D# group 0: count=1 | lds_addr | global_addr[56:0] | type=2
      v4u g0;
      g0[0] = 1u;                 // count=1, is_restore=0, gather off
      g0[1] = lds_base;           // lds_addr (bytes)
      g0[2] = (u32)ga;            // global_addr[31:0]
      g0[3] = ((u32)(ga >> 32) & 0x01FFFFFFu) | 0x80000000u;  // addr[56:32] | type=2

      // D# group 1:
      //  data_size=3 (8B), pad_enable=1, pad_interval=4 (32 DWORDs),
      //  pad_amount=1 (2 DWORDs) -> stride-17 float2 rows in LDS.
      //  tensor_dim0=65536 (all of Hft in 8B units), tensor_dim1=1,
      //  tile_dim0=4096 (32KB chunk in 8B units), tile_dim1=1,
      //  tensor_dim0_stride=65536.
      v8i g1;
      g1[0] = (int)((3u << 16) | (1u << 20) | (4u << 22) | (1u << 25));
      g1[1] = 0;                  // atomic barrier addr (unused)
      g1[2] = 0x00010001;         // tensor_dim0 hi bits (65536), tensor_dim1=1
      g1[3] = 0x10000000;         // tile_dim0 = 4096
      g1[4] = 1;                  // tile_dim1 = 1, tile_dim2 = 0
      g1[5] = 0x00010000;         // tensor_dim0_stride = 65536 (low 32)
      g1[6] = 0;
      g1[7] = 0;

      v4i g2 = {0, 0, 0, 0};      // <=2D tensor: groups 2/3 unused
      v4i g3 = {0, 0, 0, 0};
      v8i g4 = {0, 0, 0, 0, 0, 0, 0, 0};  // trailing group (6-arg builtin form)

      __builtin_amdgcn_tensor_load_to_lds(g0, g1, g2, g3, g4, 0);
      __builtin_amdgcn_s_wait_tensorcnt(0);
    }
    __syncthreads();  // publish TDM-written LDS to all waves

    // seed scaled phasors at m = m0 + tm:  (c + i s) = spw * e^{-i ang}
    float c[RANK], s[RANK];
    #pragma unroll
    for (int d = 0; d < RANK; ++d) {
      float ang = dph[d] * (float)(m0 + tm);
      float sa, ca;
      __sincosf(ang, &sa, &ca);
      c[d] = spw[d] * ca;
      s[d] = -spw[d] * sa;
    }

    for (int it = 0; it < MC / 16; ++it) {
      int ml = tm + (it << 4);
      const float2* hrow = &tile[ml * LSTR];
      float vr = 0.0f, vi = 0.0f;
      #pragma unroll
      for (int d = 0; d < RANK; ++d) {
        float2 h = hrow[d];
        vr = fmaf(c[d], h.x, vr);
        vr = fmaf(-s[d], h.y, vr);
        vi = fmaf(c[d], h.y, vi);
        vi = fmaf(s[d], h.x, vi);
        // rotate by e^{-i 16 dph}
        float nc = fmaf(c[d], cd[d],  s[d] * sd[d]);
        float ns = fmaf(s[d], cd[d], -c[d] * sd[d]);
        c[d] = nc; s[d] = ns;
      }
      float2 o; o.x = vr; o.y = vi;
      *(float2*)(out + ((size_t)n * M_COLS + (m0 + ml)) * 2) = o;
    }
  }
}

// ------------------------------------------------------------------
extern "C" void kernel_launch(void* const* d_in, const int* in_sizes, int n_in,
                              void* d_out, int out_size, void* d_ws, size_t ws_size,
                              hipStream_t stream) {
  const float* W   = (const float*)d_in[0];   // (1024, 16)
  const float* H   = (const float*)d_in[1];   // (16, 4096)
  const float* tau = (const float*)d_in[2];   // (1024, 16)
  float* out = (float*)d_out;                 // (1024, 4096) complex64 interleaved

  float* spH = (float*)((char*)d_ws + WS_SPH_OFF);
  float* Hft = (float*)((char*)d_ws + WS_HFT_OFF);

  k_softplus_h<<<(RANK * M_COLS + 255) / 256, 256, 0, stream>>>(H, spH);
  k_dft_wmma<<<M_COLS / 16, 32, 0, stream>>>(spH, Hft);
  k_main<<<N_ROWS / 16, 256, 0, stream>>>(W, tau, Hft, out);
}